// KRRGenredSolver_14731737825439
// MI455X (gfx1250) — compile-verified
//
#include <hip/hip_runtime.h>

// KRR Genred: out_i = sum_j exp(-||x_i - y_j||^2 / g) * alpha_j
// N=8192 (x rows), M=16384 (y rows), DIM=32.
//
// MI455X (gfx1250, wave32) strategy:
//  - ||x-y||^2 = x2 + y2 - 2 x.y; cross term via 8 chained
//    V_WMMA_F32_16X16X4_F32 (K=32), exact f32.
//  - K-slice remap: half-wave h owns contiguous k in [16h,16h+16) so A/B
//    fragments load as 4x global_load_b128 per lane. WMMA j uses k-pair
//    {2j,2j+1} from each half; order over K commutes.
//  - Register double-buffering: next tile's B + (c1*y2, alpha) loads issued
//    before the current tile's WMMA+exp work.
//  - exp(-sq/g) = exp2(c1*x2 + c1*y2 + (-2c1)*dot): c1*y2 pre-baked into the
//    pair table, c1*x2 pre-scaled per wave -> per element just add+fma+exp+fma
//    (no per-element multiply by c1).
//  - C/D layout (lane holds col l%16, rows v+8h) -> per-lane accumulation of
//    sum_j w*alpha, finished by 4-step __shfl_xor butterfly per half.
//  - M-sweep split 8 ways (4096 waves); deterministic partials in ws + tiny
//    reduce kernel (no float atomics).

typedef float v2f __attribute__((ext_vector_type(2)));
typedef float v8f __attribute__((ext_vector_type(8)));

#define NROWS   8192
#define MCOLS   16384
#define KDIM    32
#define SCHUNK  8                           // M-sweep split factor
#define TILES_M (MCOLS / 16)                // 1024 column tiles
#define TILES_PER_CHUNK (TILES_M / SCHUNK)  // 128

// ----------------------------------------------- (c1*y2, alpha) pair table
__global__ void __launch_bounds__(256)
krr_pair_kernel(const float* __restrict__ y, const float* __restrict__ alph,
                const float* __restrict__ g, float2* __restrict__ pair) {
    int j = blockIdx.x * blockDim.x + threadIdx.x;
    if (j < MCOLS) {
        const float c1 = -1.4426950408889634f / g[0];  // -log2(e)/g
        const float4* r = (const float4*)(y + (size_t)j * KDIM);
        float s = 0.0f;
#pragma unroll
        for (int q = 0; q < KDIM / 4; ++q) {
            float4 v = r[q];
            s += v.x * v.x + v.y * v.y + v.z * v.z + v.w * v.w;
        }
        pair[j] = make_float2(c1 * s, alph[j]);
    }
}

// ----------------------------------------------- main WMMA kernel
// grid: NROWS/16 = 512 blocks x 256 threads (8 waves).
// block b, wave w: row tile b, M-chunk w.
__global__ void __launch_bounds__(256)
krr_wmma_kernel(const float* __restrict__ x,
                const float* __restrict__ y,
                const float* __restrict__ g,
                const float2* __restrict__ pair,
                float* __restrict__ part) {
    const int lane = threadIdx.x & 31;
    const int wave = threadIdx.x >> 5;     // 0..7 -> M chunk
    const int tile = blockIdx.x;           // 0..511 row tile
    const int h    = lane >> 4;            // half-wave: owns k in [16h,16h+16)
    const int m    = lane & 15;            // row (A) / col (B) within tile
    const int row  = tile * 16 + m;

    // A fragments: contiguous 64B run x[row][16h .. 16h+16)
    const float4* xr = (const float4*)(x + (size_t)row * KDIM + 16 * h);
    float4 xq[4];
#pragma unroll
    for (int q = 0; q < 4; ++q) xq[q] = xr[q];
    v2f a[8];
#pragma unroll
    for (int q = 0; q < 4; ++q) {
        a[2 * q]     = v2f{xq[q].x, xq[q].y};
        a[2 * q + 1] = v2f{xq[q].z, xq[q].w};
    }

    // x2 of this lane's row: half-sum + partner half (xor-16)
    float x2h = 0.0f;
#pragma unroll
    for (int q = 0; q < 4; ++q)
        x2h += xq[q].x * xq[q].x + xq[q].y * xq[q].y +
               xq[q].z * xq[q].z + xq[q].w * xq[q].w;
    float x2full = x2h + __shfl_xor(x2h, 16);

    const float c1   = -1.4426950408889634f / g[0];  // -log2(e)/g
    const float m2c1 = -2.0f * c1;

    float xc[8];                            // c1 * x2 of rows v + 8h (C/D layout)
#pragma unroll
    for (int v = 0; v < 8; ++v)
        xc[v] = c1 * __shfl(x2full, v + 8 * h);

    float acc[8];
#pragma unroll
    for (int v = 0; v < 8; ++v) acc[v] = 0.0f;

    const int jt0   = wave * TILES_PER_CHUNK;
    const int jtEnd = jt0 + TILES_PER_CHUNK;

    // ---- software pipeline: preload tile jt0
    float4 bq[4];
    float2 ya;
    {
        const int col = jt0 * 16 + m;
        const float4* yr = (const float4*)(y + (size_t)col * KDIM + 16 * h);
#pragma unroll
        for (int q = 0; q < 4; ++q) bq[q] = yr[q];
        ya = pair[col];
    }

    for (int jt = jt0; jt < jtEnd; ++jt) {
        // issue next tile's loads first; power-of-2 wrap avoids cmp/cndmask
        // (last prefetch harmlessly reads the next chunk's first tile, wrapping to 0)
        const int jn = (jt + 1) & (TILES_M - 1);
        const int cn = jn * 16 + m;
        const float4* yn = (const float4*)(y + (size_t)cn * KDIM + 16 * h);
        float4 bn[4];
#pragma unroll
        for (int q = 0; q < 4; ++q) bn[q] = yn[q];
        float2 yan = pair[cn];

        // 8 chained WMMAs over the current tile's fragments
        v8f c = {};
#pragma unroll
        for (int q = 0; q < 4; ++q) {
            v2f b0 = v2f{bq[q].x, bq[q].y};
            v2f b1 = v2f{bq[q].z, bq[q].w};
            c = __builtin_amdgcn_wmma_f32_16x16x4_f32(
                    false, a[2 * q],     false, b0, (short)0, c, false, false);
            c = __builtin_amdgcn_wmma_f32_16x16x4_f32(
                    false, a[2 * q + 1], false, b1, (short)0, c, false, false);
        }

        // elementwise: e2 = c1*x2 + c1*y2 + (-2c1)*dot ; acc += exp2(e2)*alpha
        const float y2c = ya.x;               // already c1-scaled
        const float alc = ya.y;
#pragma unroll
        for (int v = 0; v < 8; ++v) {
            float t = xc[v] + y2c;
            float e = __builtin_fmaf(m2c1, c[v], t);
            acc[v] += __builtin_amdgcn_exp2f(e) * alc;
        }

        // rotate double buffer
#pragma unroll
        for (int q = 0; q < 4; ++q) bq[q] = bn[q];
        ya = yan;
    }

    // column reduction: butterfly over the 16 lanes of each half-wave
#pragma unroll
    for (int v = 0; v < 8; ++v) {
#pragma unroll
        for (int msk = 1; msk < 16; msk <<= 1)
            acc[v] += __shfl_xor(acc[v], msk);
    }
    if (m == 0) {
        // lane 0 -> rows 0..7, lane 16 -> rows 8..15 of this tile
        float* p = part + (size_t)wave * NROWS + tile * 16 + 8 * h;
#pragma unroll
        for (int v = 0; v < 8; ++v) p[v] = acc[v];
    }
}

// ----------------------------------------------- final reduce over chunks
__global__ void __launch_bounds__(256)
krr_reduce_kernel(const float* __restrict__ part, float* __restrict__ out) {
    int i = blockIdx.x * blockDim.x + threadIdx.x;
    if (i < NROWS) {
        float s = 0.0f;
#pragma unroll
        for (int sC = 0; sC < SCHUNK; ++sC)
            s += part[(size_t)sC * NROWS + i];
        out[i] = s;
    }
}

extern "C" void kernel_launch(void* const* d_in, const int* in_sizes, int n_in,
                              void* d_out, int out_size, void* d_ws, size_t ws_size,
                              hipStream_t stream) {
    const float* x    = (const float*)d_in[0];   // [N, 32]
    const float* y    = (const float*)d_in[1];   // [M, 32]
    const float* alph = (const float*)d_in[2];   // [M, 1]
    const float* g    = (const float*)d_in[3];   // [1]
    float* out = (float*)d_out;                  // [N, 1]

    // ws layout: [0, M) float2 (c1*y2, alpha) ; then SCHUNK*N float partials
    float2* pair = (float2*)d_ws;
    float*  part = (float*)(pair + MCOLS);

    krr_pair_kernel<<<MCOLS / 256, 256, 0, stream>>>(y, alph, g, pair);
    krr_wmma_kernel<<<NROWS / 16, 256, 0, stream>>>(x, y, g, pair, part);
    krr_reduce_kernel<<<NROWS / 256, 256, 0, stream>>>(part, out);
}